// model_1_76166950028523
// MI455X (gfx1250) — compile-verified
//
#include <hip/hip_runtime.h>
#include <hip/hip_bf16.h>

// ---------------------------------------------------------------------------
// Dependency-parser pipeline for MI455X (gfx1250, wave32):
//   embed -> [gemm gx (WMMA f32, 16x64 tiles/wave, packed-B)]
//         -> persistent LSTM recurrence (x2 dirs) -> layer1 same
//         -> hA/hB projections (WMMA f32) -> pairwise tanh scorer
// ---------------------------------------------------------------------------

#define S_LEN   768
#define V_WORD  30000
#define V_POS   64
#define WEMB    100
#define PEMB    28
#define EMB     128
#define HDIM    256
#define G4H     1024   // 4*H
#define MLP     128

typedef __attribute__((ext_vector_type(2))) float v2f;
typedef __attribute__((ext_vector_type(8))) float v8f;

__device__ __forceinline__ float fast_sigmoid(float x) {
    return 1.0f / (1.0f + __expf(-x));
}
__device__ __forceinline__ float fast_tanh(float x) {
    float c = fminf(fmaxf(x, -9.0f), 9.0f);
    float e = __expf(2.0f * c);
    return (e - 1.0f) / (e + 1.0f);
}

// ---------------------------------------------------------------------------
// Embedding gather: x[s, 0:100] = W_word[widx[s]], x[s, 100:128] = W_pos[pidx[s]]
// ---------------------------------------------------------------------------
__global__ void embed_kernel(const int* __restrict__ widx, const int* __restrict__ pidx,
                             const float* __restrict__ Wword, const float* __restrict__ Wpos,
                             float* __restrict__ x) {
    int s = blockIdx.x;
    int t = threadIdx.x;           // 0..127
    float v;
    if (t < WEMB) v = Wword[(size_t)widx[s] * WEMB + t];
    else          v = Wpos[(size_t)pidx[s] * PEMB + (t - WEMB)];
    x[(size_t)s * EMB + t] = v;
}

// ---------------------------------------------------------------------------
// Plain sub-matrix transpose (used for Whh^T in the recurrence):
//   dst[c*R + r] = src[r*Ctot + coff + c]
// ---------------------------------------------------------------------------
__global__ void transpose_sub(const float* __restrict__ src, float* __restrict__ dst,
                              int R, int Ctot, int coff, int Ccnt) {
    int idx = blockIdx.x * blockDim.x + threadIdx.x;
    int total = R * Ccnt;
    if (idx >= total) return;
    int r = idx / Ccnt;
    int c = idx - r * Ccnt;
    dst[(size_t)c * R + r] = src[(size_t)r * Ctot + coff + c];
}

// ---------------------------------------------------------------------------
// Pack weight W [Nout x Ctot] (row-major, taking columns [coff, coff+Kcnt))
// into WMMA-fragment order:  BP[((k>>2)*Nout + n)*4 + (k&3)] = W[n][coff+k]
// Each lane's (B[k0+kb][n], B[k0+kb+1][n]) pair becomes contiguous -> b64 load.
// ---------------------------------------------------------------------------
__global__ void pack_bt(const float* __restrict__ src, float* __restrict__ dst,
                        int Nout, int Ctot, int coff, int Kcnt) {
    int idx = blockIdx.x * blockDim.x + threadIdx.x;
    int total = Nout * Kcnt;
    if (idx >= total) return;
    int n = idx / Kcnt;
    int k = idx - n * Kcnt;
    dst[((size_t)(k >> 2) * Nout + n) * 4 + (k & 3)] = src[(size_t)n * Ctot + coff + k];
}

// ---------------------------------------------------------------------------
// WMMA f32 GEMM:  C[M,N] = A[M,K] @ B   (B supplied packed as BP, see pack_bt)
// One wave computes a 16x64 output tile (4 sub-tiles / 4 accumulators) via
// V_WMMA_F32_16X16X4_F32; the A fragment is reused across 4 WMMAs per k-step.
//   A-fragment (ISA 7.12.2, f32 16x4):  lane l: m = l&15, kb = (l>>4)*2
//       vgpr0 = A[m][k0+kb], vgpr1 = A[m][k0+kb+1]
//   B-fragment (4x16, row-striped):     lane l: n = l&15
//       vgpr0 = B[k0+kb][n], vgpr1 = B[k0+kb+1][n]
//   C/D: vgpr v, lane l: C[v + (l>>4)*8][l&15]
// Optional bias0/bias1 (length N) added in the epilogue.
// ---------------------------------------------------------------------------
__global__ void __launch_bounds__(32)
gemm_wmma_f32(const float* __restrict__ A, const float* __restrict__ BP,
              const float* __restrict__ bias0, const float* __restrict__ bias1,
              float* __restrict__ C, int M, int N, int K) {
    const int nt = blockIdx.x;     // 64-wide N tile
    const int mt = blockIdx.y;     // 16-high M tile
    const int l  = threadIdx.x;    // 0..31, EXEC all-ones
    const int fr   = l & 15;       // m-fragment == n-fragment index
    const int half = l >> 4;
    const int kb   = half * 2;

    const float* Arow  = A + (size_t)(mt * 16 + fr) * K + kb;
    const int    nbase = nt * 64;

    v8f acc[4] = {{}, {}, {}, {}};
    for (int k0 = 0; k0 < K; k0 += 4) {
        v2f a;
        a.x = Arow[k0];
        a.y = Arow[k0 + 1];
        // packed B: fragment base for (k0, n) is ((k0/4)*N + n)*4 + kb
        const float* bp = BP + ((size_t)(k0 >> 2) * N + nbase + fr) * 4 + kb;
        v2f b0, b1, b2, b3;
        b0.x = bp[0];       b0.y = bp[1];        // n-subtile 0
        b1.x = bp[16 * 4];  b1.y = bp[16 * 4 + 1];
        b2.x = bp[32 * 4];  b2.y = bp[32 * 4 + 1];
        b3.x = bp[48 * 4];  b3.y = bp[48 * 4 + 1];
        acc[0] = __builtin_amdgcn_wmma_f32_16x16x4_f32(false, a, false, b0, (short)0, acc[0], false, false);
        acc[1] = __builtin_amdgcn_wmma_f32_16x16x4_f32(false, a, false, b1, (short)0, acc[1], false, false);
        acc[2] = __builtin_amdgcn_wmma_f32_16x16x4_f32(false, a, false, b2, (short)0, acc[2], false, false);
        acc[3] = __builtin_amdgcn_wmma_f32_16x16x4_f32(false, a, false, b3, (short)0, acc[3], false, false);
    }

#pragma unroll
    for (int u = 0; u < 4; ++u) {
        const int g = nbase + u * 16 + fr;     // output column
        float bs = 0.0f;
        if (bias0) bs += bias0[g];
        if (bias1) bs += bias1[g];
#pragma unroll
        for (int v = 0; v < 8; ++v) {
            int row = v + half * 8;
            C[(size_t)(mt * 16 + row) * N + g] = acc[u][v] + bs;
        }
    }
}

// ---------------------------------------------------------------------------
// Persistent bidirectional LSTM layer. grid = 2 (dir), block = 1024 threads.
// Each thread owns one of the 4H=1024 gate pre-activations per step:
//   g[tid] = gx[t, tid] + sum_j h[j] * WhhT[j, tid]     (WhhT is H x 4H)
// Threads 0..255 then apply the gate nonlinearity (i,f,g,o order).
// hout layout: [S][2H], forward dir -> cols 0..255, backward -> 256..511
// ---------------------------------------------------------------------------
__global__ void lstm_layer(const float* __restrict__ gx_f, const float* __restrict__ gx_b,
                           const float* __restrict__ WhhT_f, const float* __restrict__ WhhT_b,
                           float* __restrict__ hout) {
    const int dir = blockIdx.x;                 // 0 = forward, 1 = backward
    const float* gx   = dir ? gx_b   : gx_f;
    const float* WhhT = dir ? WhhT_b : WhhT_f;

    __shared__ float h_s[HDIM];
    __shared__ float c_s[HDIM];
    __shared__ float g_s[G4H];

    const int tid = threadIdx.x;                // 0..1023 (gate index)
    if (tid < HDIM) { h_s[tid] = 0.0f; c_s[tid] = 0.0f; }
    __syncthreads();

    for (int step = 0; step < S_LEN; ++step) {
        const int t = dir ? (S_LEN - 1 - step) : step;
        float acc = gx[(size_t)t * G4H + tid];
#pragma unroll 8
        for (int j = 0; j < HDIM; ++j) {
            acc += h_s[j] * WhhT[(size_t)j * G4H + tid];
        }
        g_s[tid] = acc;
        __syncthreads();
        if (tid < HDIM) {
            float gi = g_s[tid];
            float gf = g_s[HDIM + tid];
            float gc = g_s[2 * HDIM + tid];
            float go = g_s[3 * HDIM + tid];
            float c  = fast_sigmoid(gf) * c_s[tid] + fast_sigmoid(gi) * fast_tanh(gc);
            float h  = fast_sigmoid(go) * fast_tanh(c);
            c_s[tid] = c;
            h_s[tid] = h;
            hout[(size_t)t * (2 * HDIM) + dir * HDIM + tid] = h;
        }
        __syncthreads();
    }
}

// ---------------------------------------------------------------------------
// Pairwise edge scorer. grid (S/16, S/16), block 256 = 16x16 (i,j) pairs.
//   scores[i,j] = sum_k tanh(hA[i,k] + hB[j,k]) * W2[k] + b2   (b1 baked in hA)
//   diagonal forced to 0.
// ---------------------------------------------------------------------------
__global__ void pairwise_kernel(const float* __restrict__ hA, const float* __restrict__ hB,
                                const float* __restrict__ W2, const float* __restrict__ b2,
                                float* __restrict__ out) {
    __shared__ float sA[16][MLP];
    __shared__ float sB[16][MLP];
    __shared__ float sW[MLP];

    const int i0 = blockIdx.y * 16;
    const int j0 = blockIdx.x * 16;
    const int tid = threadIdx.x;                // 0..255

    for (int idx = tid; idx < 16 * MLP; idx += 256) {
        int r = idx >> 7, k = idx & (MLP - 1);
        sA[r][k] = hA[(size_t)(i0 + r) * MLP + k];
        sB[r][k] = hB[(size_t)(j0 + r) * MLP + k];
    }
    if (tid < MLP) sW[tid] = W2[tid];
    __syncthreads();

    const int ti = tid >> 4;
    const int tj = tid & 15;
    const int i = i0 + ti;
    const int j = j0 + tj;

    float acc = 0.0f;
#pragma unroll 4
    for (int k = 0; k < MLP; ++k) {
        acc += sW[k] * fast_tanh(sA[ti][k] + sB[tj][k]);
    }
    float r = acc + b2[0];
    out[(size_t)i * S_LEN + j] = (i == j) ? 0.0f : r;
}

// ---------------------------------------------------------------------------
// Launcher
// ---------------------------------------------------------------------------
extern "C" void kernel_launch(void* const* d_in, const int* in_sizes, int n_in,
                              void* d_out, int out_size, void* d_ws, size_t ws_size,
                              hipStream_t stream) {
    const int*   widx   = (const int*)d_in[0];
    const int*   pidx   = (const int*)d_in[1];
    // d_in[2] = sentence_len (unused, S fixed)
    const float* Wword  = (const float*)d_in[3];
    const float* Wpos   = (const float*)d_in[4];
    const float* Wih_l0f = (const float*)d_in[5];
    const float* Whh_l0f = (const float*)d_in[6];
    const float* bih_l0f = (const float*)d_in[7];
    const float* bhh_l0f = (const float*)d_in[8];
    const float* Wih_l0b = (const float*)d_in[9];
    const float* Whh_l0b = (const float*)d_in[10];
    const float* bih_l0b = (const float*)d_in[11];
    const float* bhh_l0b = (const float*)d_in[12];
    const float* Wih_l1f = (const float*)d_in[13];
    const float* Whh_l1f = (const float*)d_in[14];
    const float* bih_l1f = (const float*)d_in[15];
    const float* bhh_l1f = (const float*)d_in[16];
    const float* Wih_l1b = (const float*)d_in[17];
    const float* Whh_l1b = (const float*)d_in[18];
    const float* bih_l1b = (const float*)d_in[19];
    const float* bhh_l1b = (const float*)d_in[20];
    const float* W1 = (const float*)d_in[21];
    const float* b1 = (const float*)d_in[22];
    const float* W2 = (const float*)d_in[23];
    const float* b2 = (const float*)d_in[24];
    float* out = (float*)d_out;

    // ---- workspace carve-up (floats) ----
    float* ws = (float*)d_ws;
    float* x      = ws;                          // 768*128
    float* gxf    = x      + S_LEN * EMB;        // 768*1024
    float* gxb    = gxf    + S_LEN * G4H;        // 768*1024
    float* h0     = gxb    + S_LEN * G4H;        // 768*512
    float* h1     = h0     + S_LEN * 2 * HDIM;   // 768*512
    float* hA     = h1     + S_LEN * 2 * HDIM;   // 768*128
    float* hB     = hA     + S_LEN * MLP;        // 768*128
    float* BP0f   = hB     + S_LEN * MLP;        // packed Wih_l0f: 128*1024
    float* BP0b   = BP0f   + EMB * G4H;
    float* BP1f   = BP0b   + EMB * G4H;          // packed Wih_l1f: 512*1024
    float* BP1b   = BP1f   + 2 * HDIM * G4H;
    float* WhhT0f = BP1b   + 2 * HDIM * G4H;     // 256*1024
    float* WhhT0b = WhhT0f + HDIM * G4H;
    float* WhhT1f = WhhT0b + HDIM * G4H;
    float* WhhT1b = WhhT1f + HDIM * G4H;
    float* BPA    = WhhT1b + HDIM * G4H;         // packed W1[:, :512]: 512*128
    float* BPB    = BPA    + 2 * HDIM * MLP;     // packed W1[:, 512:]: 512*128

    // ---- weight packing / transposes ----
    auto tgrid = [](int total) { return dim3((total + 255) / 256); };
    pack_bt<<<tgrid(G4H * EMB),      256, 0, stream>>>(Wih_l0f, BP0f, G4H, EMB,      0,      EMB);
    pack_bt<<<tgrid(G4H * EMB),      256, 0, stream>>>(Wih_l0b, BP0b, G4H, EMB,      0,      EMB);
    pack_bt<<<tgrid(G4H * 2 * HDIM), 256, 0, stream>>>(Wih_l1f, BP1f, G4H, 2 * HDIM, 0, 2 * HDIM);
    pack_bt<<<tgrid(G4H * 2 * HDIM), 256, 0, stream>>>(Wih_l1b, BP1b, G4H, 2 * HDIM, 0, 2 * HDIM);
    pack_bt<<<tgrid(MLP * 2 * HDIM), 256, 0, stream>>>(W1, BPA, MLP, G4H,        0, 2 * HDIM);
    pack_bt<<<tgrid(MLP * 2 * HDIM), 256, 0, stream>>>(W1, BPB, MLP, G4H, 2 * HDIM, 2 * HDIM);
    transpose_sub<<<tgrid(G4H * HDIM), 256, 0, stream>>>(Whh_l0f, WhhT0f, G4H, HDIM, 0, HDIM);
    transpose_sub<<<tgrid(G4H * HDIM), 256, 0, stream>>>(Whh_l0b, WhhT0b, G4H, HDIM, 0, HDIM);
    transpose_sub<<<tgrid(G4H * HDIM), 256, 0, stream>>>(Whh_l1f, WhhT1f, G4H, HDIM, 0, HDIM);
    transpose_sub<<<tgrid(G4H * HDIM), 256, 0, stream>>>(Whh_l1b, WhhT1b, G4H, HDIM, 0, HDIM);

    // ---- embeddings ----
    embed_kernel<<<S_LEN, EMB, 0, stream>>>(widx, pidx, Wword, Wpos, x);

    // ---- layer 0: gx GEMMs (WMMA) + recurrence ----
    {
        dim3 grid(G4H / 64, S_LEN / 16);
        gemm_wmma_f32<<<grid, 32, 0, stream>>>(x, BP0f, bih_l0f, bhh_l0f, gxf, S_LEN, G4H, EMB);
        gemm_wmma_f32<<<grid, 32, 0, stream>>>(x, BP0b, bih_l0b, bhh_l0b, gxb, S_LEN, G4H, EMB);
    }
    lstm_layer<<<2, G4H, 0, stream>>>(gxf, gxb, WhhT0f, WhhT0b, h0);

    // ---- layer 1 ----
    {
        dim3 grid(G4H / 64, S_LEN / 16);
        gemm_wmma_f32<<<grid, 32, 0, stream>>>(h0, BP1f, bih_l1f, bhh_l1f, gxf, S_LEN, G4H, 2 * HDIM);
        gemm_wmma_f32<<<grid, 32, 0, stream>>>(h0, BP1b, bih_l1b, bhh_l1b, gxb, S_LEN, G4H, 2 * HDIM);
    }
    lstm_layer<<<2, G4H, 0, stream>>>(gxf, gxb, WhhT1f, WhhT1b, h1);

    // ---- scorer projections (WMMA); b1 folded into hA ----
    {
        dim3 grid(MLP / 64, S_LEN / 16);
        gemm_wmma_f32<<<grid, 32, 0, stream>>>(h1, BPA, b1, (const float*)nullptr, hA, S_LEN, MLP, 2 * HDIM);
        gemm_wmma_f32<<<grid, 32, 0, stream>>>(h1, BPB, (const float*)nullptr, (const float*)nullptr, hB, S_LEN, MLP, 2 * HDIM);
    }

    // ---- pairwise scores ----
    {
        dim3 grid(S_LEN / 16, S_LEN / 16);
        pairwise_kernel<<<grid, 256, 0, stream>>>(hA, hB, W2, b2, out);
    }
}